// Loss_19499151524481
// MI455X (gfx1250) — compile-verified
//
#include <hip/hip_runtime.h>

// Cross-entropy loss, [B=8192, C=32000] fp32, out = sum(logZ_row - x[row,tgt]) / 3.
// Bandwidth-bound: 1.05 GB single-pass stream @ 23.3 TB/s => ~45us floor.

typedef float f4  __attribute__((ext_vector_type(4)));
typedef float v2f __attribute__((ext_vector_type(2)));
typedef float v8f __attribute__((ext_vector_type(8)));

#define THREADS 256
#define WAVES   (THREADS / 32)
#define VPT     8                    // float4 per thread per chunk
#define CHUNKV  (THREADS * VPT)      // 2048 float4 = 8192 floats per chunk

// One block per row. Chunked two-pass logsumexp with register-resident tiles:
// data read from HBM exactly once (NT-hinted b128 loads), merged with the
// streaming (M,S) logsumexp combine across chunks.
__global__ __launch_bounds__(THREADS)
void row_logsumexp_pick(const float* __restrict__ x,
                        const int*  __restrict__ tgt,
                        float*      __restrict__ ws,
                        int C)
{
    const int row  = blockIdx.x;
    const f4* row4 = (const f4*)(x + (size_t)row * C);
    const int nv4    = C >> 2;
    const int nchunk = (nv4 + CHUNKV - 1) / CHUNKV;

    __shared__ float redM[WAVES];
    __shared__ float redS[WAVES];

    const int t    = threadIdx.x;
    const int lane = t & 31;
    const int wid  = t >> 5;

    float M = -3.0e38f;   // running max
    float S = 0.0f;       // running sum of exp(x - M)

    for (int ch = 0; ch < nchunk; ++ch) {
        // Load chunk into registers (coalesced 16B/lane, non-temporal: no reuse)
        f4 v[VPT];
        const int base = ch * CHUNKV + t;
#pragma unroll
        for (int k = 0; k < VPT; ++k) {
            int idx = base + k * THREADS;
            if (idx < nv4) {
                v[k] = __builtin_nontemporal_load(&row4[idx]);
            } else {
                f4 s; s.x = s.y = s.z = s.w = -3.0e38f; v[k] = s;
            }
        }

        // Pass 1: chunk max (registers -> wave32 shuffle -> LDS across 8 waves)
        float lm = -3.0e38f;
#pragma unroll
        for (int k = 0; k < VPT; ++k)
            lm = fmaxf(lm, fmaxf(fmaxf(v[k].x, v[k].y), fmaxf(v[k].z, v[k].w)));
#pragma unroll
        for (int m = 16; m >= 1; m >>= 1)
            lm = fmaxf(lm, __shfl_xor(lm, m, 32));
        if (lane == 0) redM[wid] = lm;
        __syncthreads();
        float Mc = redM[0];
#pragma unroll
        for (int w = 1; w < WAVES; ++w) Mc = fmaxf(Mc, redM[w]);

        // Pass 2: chunk sum of exp(x - Mc)  (sentinel lanes underflow to 0)
        float ls = 0.0f;
#pragma unroll
        for (int k = 0; k < VPT; ++k) {
            ls += __expf(v[k].x - Mc);
            ls += __expf(v[k].y - Mc);
            ls += __expf(v[k].z - Mc);
            ls += __expf(v[k].w - Mc);
        }
#pragma unroll
        for (int m = 16; m >= 1; m >>= 1)
            ls += __shfl_xor(ls, m, 32);
        if (lane == 0) redS[wid] = ls;
        __syncthreads();
        float Sc = 0.0f;
#pragma unroll
        for (int w = 0; w < WAVES; ++w) Sc += redS[w];

        // Streaming logsumexp merge (computed redundantly by all threads)
        float nM = fmaxf(M, Mc);
        S = S * __expf(M - nM) + Sc * __expf(Mc - nM);
        M = nM;
        // No extra barrier needed: next chunk's LDS writes are ordered behind
        // this chunk's barriers relative to all reads above.
    }

    if (t == 0) {
        int   ti = tgt[row];
        float xt = x[(size_t)row * C + ti];
        ws[row] = (M + __logf(S)) - xt;   // = -(logp at target)
    }
}

// Deterministic final reduction over B per-row values.
// 256 threads so the loads have memory-level parallelism (8 independent
// float4 loads in flight per thread instead of a serial dependent chain).
// Wave32 shuffle reduce -> 8 per-wave partials in LDS -> wave 0 folds them
// with V_WMMA_F32_16X16X4_F32 (A holds lane partials, B = ones =>
// D[m][n] = p_m + p_{m+16}; summing D's 8 VGPRs + lane0/lane16 gives the sum).
__global__ __launch_bounds__(THREADS)
void final_reduce(const float* __restrict__ ws, float* __restrict__ out,
                  int B, float inv_div)
{
    const int t    = threadIdx.x;
    const int lane = t & 31;
    const int wid  = t >> 5;
    const int nv4  = B >> 2;                 // B is a multiple of 4 (8192)
    const f4* ws4  = (const f4*)ws;

    __shared__ float red[WAVES];

    // Gather: fixed per-thread order => deterministic; loads all independent.
    f4 acc; acc.x = acc.y = acc.z = acc.w = 0.0f;
#pragma unroll 8
    for (int i = t; i < nv4; i += THREADS) {
        f4 v = ws4[i];
        acc.x += v.x; acc.y += v.y; acc.z += v.z; acc.w += v.w;
    }
    float p = (acc.x + acc.y) + (acc.z + acc.w);

#pragma unroll
    for (int m = 16; m >= 1; m >>= 1)
        p += __shfl_xor(p, m, 32);
    if (lane == 0) red[wid] = p;
    __syncthreads();

    if (wid == 0) {  // uniform per-wave branch: wave 0 has EXEC all ones
        float q = (lane < WAVES) ? red[lane] : 0.0f;
#if __has_builtin(__builtin_amdgcn_wmma_f32_16x16x4_f32)
        v2f a; a[0] = q;    a[1] = 0.0f;   // A: 16x4, partials in K=0 / K=2
        v2f b; b[0] = 1.0f; b[1] = 1.0f;   // B: 4x16 all-ones
        v8f c = {};
        v8f d = __builtin_amdgcn_wmma_f32_16x16x4_f32(
            /*neg_a=*/false, a, /*neg_b=*/false, b,
            /*c_mod=*/(short)0, c, /*reuse_a=*/false, /*reuse_b=*/false);
        float tsv = d[0] + d[1] + d[2] + d[3] + d[4] + d[5] + d[6] + d[7];
        float tot = __shfl(tsv, 0, 32) + __shfl(tsv, 16, 32);
#else
        float tot = q;
#pragma unroll
        for (int m = 16; m >= 1; m >>= 1) tot += __shfl_xor(tot, m, 32);
#endif
        if (lane == 0) out[0] = tot * inv_div;
    }
}

extern "C" void kernel_launch(void* const* d_in, const int* in_sizes, int n_in,
                              void* d_out, int out_size, void* d_ws, size_t ws_size,
                              hipStream_t stream)
{
    const float* x   = (const float*)d_in[0];   // [B, C] fp32 logits
    const int*   tgt = (const int*)d_in[1];     // [B] targets
    const int B = in_sizes[1];
    const int C = in_sizes[0] / B;

    float* ws = (float*)d_ws;                   // B floats of per-row loss

    row_logsumexp_pick<<<B, THREADS, 0, stream>>>(x, tgt, ws, C);
    final_reduce<<<1, THREADS, 0, stream>>>(ws, (float*)d_out, B, 1.0f / 3.0f);
}